// _ROISampler_6743098655767
// MI455X (gfx1250) — compile-verified
//
#include <hip/hip_runtime.h>
#include <hip/hip_bf16.h>

// ---------------- problem constants (from reference) ----------------
#define BATCH        32
#define NUM_ROIS     8192
#define NUM_GT       128
#define NTOT         (NUM_ROIS + NUM_GT)   // 8320
#define NUM_SAMPLED  512
#define K_POS        128                   // 512 * 0.25
#define K_NEG        384                   // 512 - 128
#define SEL_THREADS  1024
#define CHUNK        ((NTOT + SEL_THREADS - 1) / SEL_THREADS)  // 9

// ---------------- CDNA5 TDM (Tensor Data Mover) helpers ----------------
typedef unsigned int u32x4 __attribute__((ext_vector_type(4)));
typedef int          i32x4 __attribute__((ext_vector_type(4)));
typedef int          i32x8 __attribute__((ext_vector_type(8)));

#if defined(__AMDGCN__) && (__has_builtin(__builtin_amdgcn_tensor_load_to_lds_d2) || \
                            __has_builtin(__builtin_amdgcn_tensor_load_to_lds))
#define USE_TDM 1
#else
#define USE_TDM 0
#endif

#if USE_TDM
// 1-D contiguous DMA of `ndw` dwords from global memory into LDS.
// Builds a D# per CDNA5 ISA:
//   group0: count=1 valid, lds_addr, 57-bit global addr, type=2 ("image")
//   group1: data_size=4B, tensor_dim0=tile_dim0=ndw, tensor_dim1=1, stride0=ndw
__device__ __forceinline__ void tdm_load_dwords_to_lds(void* lds_dst, const void* gsrc,
                                                       unsigned ndw) {
  // Generic pointers to LDS carry the LDS byte offset in addr[31:0] (aperture in
  // the high dword), so truncation yields D#.lds_addr.
  unsigned lds_off = (unsigned)(unsigned long long)lds_dst;
  unsigned long long ga = (unsigned long long)gsrc;
  u32x4 g0;
  g0[0] = 1u;                                              // count=1 (valid user D#)
  g0[1] = lds_off;                                         // lds_addr (bytes)
  g0[2] = (unsigned)(ga & 0xFFFFFFFFull);                  // global_addr[31:0]
  g0[3] = (unsigned)((ga >> 32) & 0x1FFFFFFull)            // global_addr[56:32]
        | 0x80000000u;                                     // type=2 in bits[127:126]
  i32x8 g1;
  g1[0] = (int)(2u << 16);                                 // data_size=2 (4 bytes), no mask/pad
  g1[1] = (int)((ndw & 0xFFFFu) << 16);                    // tensor_dim0[15:0]  (bits 63:48)
  g1[2] = (int)(((ndw >> 16) & 0xFFFFu) | (1u << 16));     // tensor_dim0[31:16], tensor_dim1=1
  g1[3] = (int)((ndw & 0xFFFFu) << 16);                    // tile_dim0 (bits 127:112)
  g1[4] = 0;                                               // tile_dim1/tile_dim2 unused
  g1[5] = (int)ndw;                                        // tensor_dim0_stride[31:0]
  g1[6] = 0;                                               // stride0 hi / stride1 lo
  g1[7] = 0;                                               // stride1 hi
#if __has_builtin(__builtin_amdgcn_tensor_load_to_lds_d2)
  __builtin_amdgcn_tensor_load_to_lds_d2(g0, g1, 0);
#else
  i32x4 z4 = (i32x4)0;
#if defined(__clang_major__) && __clang_major__ >= 23
  i32x8 z8 = (i32x8)0;
  __builtin_amdgcn_tensor_load_to_lds(g0, g1, z4, z4, z8, 0);
#else
  __builtin_amdgcn_tensor_load_to_lds(g0, g1, z4, z4, 0);
#endif
#endif
}
#endif  // USE_TDM

__device__ __forceinline__ void wait_tensorcnt0() {
#if defined(__AMDGCN__)
#if __has_builtin(__builtin_amdgcn_s_wait_tensorcnt)
  __builtin_amdgcn_s_wait_tensorcnt(0);
#else
  asm volatile("s_wait_tensorcnt 0x0" ::: "memory");
#endif
#endif
}

// Stage `ndw` dwords of gt data into LDS: TDM issued by wave 0, others fall
// through to s_wait (their TENSORcnt is 0) and the workgroup barrier.
__device__ __forceinline__ void stage_to_lds(float* lds_dst, const float* gsrc,
                                             unsigned ndw) {
#if USE_TDM
  if (threadIdx.x < 32) {
    tdm_load_dwords_to_lds((void*)lds_dst, (const void*)gsrc, ndw);
  }
#else
  for (unsigned t = threadIdx.x; t < ndw; t += blockDim.x) lds_dst[t] = gsrc[t];
#endif
}

// ---------------- kernel 1: IoU argmax / match flags ----------------
// match[b*NTOT+i] = argmax_col | (pos<<8) | (neg<<9)
__global__ __launch_bounds__(256) void match_kernel(const float* __restrict__ rois,
                                                    const float* __restrict__ gt_boxes,
                                                    int* __restrict__ match) {
  const int b = blockIdx.y;
  __shared__ float sgt[NUM_GT * 4];
  stage_to_lds(sgt, gt_boxes + (size_t)b * NUM_GT * 4, NUM_GT * 4);

  const int i = blockIdx.x * 256 + threadIdx.x;
  float a0 = 0.f, a1 = 0.f, a2 = 0.f, a3 = 0.f;
  bool from_rois = (i < NUM_ROIS);
  if (from_rois) {
    const float4* r4 = reinterpret_cast<const float4*>(rois);
    __builtin_prefetch(&r4[(size_t)b * NUM_ROIS + i], 0, 1);  // overlap with TDM
    float4 v = r4[(size_t)b * NUM_ROIS + i];
    a0 = v.x; a1 = v.y; a2 = v.z; a3 = v.w;
  }
  wait_tensorcnt0();
  __syncthreads();
  if (i >= NTOT) return;
  if (!from_rois) {
    int g = i - NUM_ROIS;
    a0 = sgt[g * 4 + 0]; a1 = sgt[g * 4 + 1]; a2 = sgt[g * 4 + 2]; a3 = sgt[g * 4 + 3];
  }
  const bool mask_a = (a0 == -1.f) && (a1 == -1.f) && (a2 == -1.f) && (a3 == -1.f);
  const float area_a = (a2 - a0) * (a3 - a1);

  float best = -__builtin_huge_valf();
  int bestj = 0;
  for (int j = 0; j < NUM_GT; ++j) {
    float b0 = sgt[j * 4 + 0], b1 = sgt[j * 4 + 1];
    float b2 = sgt[j * 4 + 2], b3 = sgt[j * 4 + 3];
    float iy = fmaxf(fminf(a2, b2) - fmaxf(a0, b0), 0.f);
    float ix = fmaxf(fminf(a3, b3) - fmaxf(a1, b1), 0.f);
    float inter = iy * ix;
    float area_b = (b2 - b0) * (b3 - b1);
    float uni = area_a + area_b - inter;
    float iou = inter / fmaxf(uni, 1e-8f);
    bool mask_b = (b0 == -1.f) && (b1 == -1.f) && (b2 == -1.f) && (b3 == -1.f);
    if (mask_a || mask_b) iou = -1.f;
    if (iou > best) { best = iou; bestj = j; }   // first-occurrence argmax
  }
  int pos = (best >= 0.5f) ? 1 : 0;
  int neg = (!pos && best >= 0.0f) ? 1 : 0;
  match[(size_t)b * NTOT + i] = bestj | (pos << 8) | (neg << 9);
}

// ---------------- kernel 2: exact balanced sampling ----------------
__device__ int block_count_ge(const unsigned* keys, unsigned T, int* cnt) {
  const int tid = threadIdx.x;
  if (tid == 0) *cnt = 0;
  __syncthreads();
  int local = 0;
  for (int i = tid; i < NTOT; i += SEL_THREADS) local += (keys[i] >= T) ? 1 : 0;
  if (local) atomicAdd(cnt, local);
  __syncthreads();
  int r = *cnt;
  __syncthreads();
  return r;
}

// Mark top-K keys (desc value, ties -> ascending index) among flagged entries.
// Flagged keys carry bit31; unflagged entries are 0.  Reproduces jax.lax.top_k
// semantics exactly (stable, bit-level comparison of non-negative floats).
__device__ void select_topk(const unsigned* keys, unsigned char* ind, int K,
                            int* cnt, int* scan) {
  const int tid = threadIdx.x;
  int P = block_count_ge(keys, 0x80000000u, cnt);
  if (P <= K) {  // fewer candidates than K: take them all
    for (int i = tid; i < NTOT; i += SEL_THREADS)
      if (keys[i] >= 0x80000000u) ind[i] = 1;
    __syncthreads();
    return;
  }
  // largest T with count(key >= T) >= K  ->  T is the K-th largest key
  unsigned lo = 0x80000000u, hi = 0xFFFFFFFFu;
  while (lo < hi) {
    unsigned mid = lo + ((hi - lo + 1u) >> 1);
    int c = block_count_ge(keys, mid, cnt);
    if (c >= K) lo = mid; else hi = mid - 1u;
  }
  const unsigned T = lo;
  const int g = (T == 0xFFFFFFFFu) ? 0 : block_count_ge(keys, T + 1u, cnt);
  for (int i = tid; i < NTOT; i += SEL_THREADS)
    if (keys[i] > T) ind[i] = 1;
  // first (K-g) entries equal to T, ascending index: chunked block scan
  const int need = K - g;
  int lo_i = tid * CHUNK;
  int hi_i = lo_i + CHUNK; if (hi_i > NTOT) hi_i = NTOT;
  int local = 0;
  for (int i = lo_i; i < hi_i; ++i) local += (keys[i] == T) ? 1 : 0;
  scan[tid] = local;
  __syncthreads();
  for (int off = 1; off < SEL_THREADS; off <<= 1) {
    int v = (tid >= off) ? scan[tid - off] : 0;
    __syncthreads();
    scan[tid] += v;
    __syncthreads();
  }
  int r = scan[tid] - local;  // exclusive prefix of ==T before my chunk
  for (int i = lo_i; i < hi_i; ++i) {
    if (keys[i] == T) { if (r < need) ind[i] = 1; ++r; }
  }
  __syncthreads();
}

__global__ __launch_bounds__(SEL_THREADS) void sample_kernel(
    const float* __restrict__ noise, const int* __restrict__ match,
    int* __restrict__ sampled_idx, int* __restrict__ num_ones) {
  const int b = blockIdx.x;
  const int tid = threadIdx.x;
  __shared__ unsigned keys[NTOT];
  __shared__ unsigned char ind[NTOT];
  __shared__ int scan[SEL_THREADS];
  __shared__ int cnt;

  for (int i = tid; i < NTOT; i += SEL_THREADS) ind[i] = 0;

  // positives, noise channel 0 (sign bit flags candidacy; noise in [0,1) so bit31 free)
  for (int i = tid; i < NTOT; i += SEL_THREADS) {
    int m = match[(size_t)b * NTOT + i];
    float n0 = noise[((size_t)b * NTOT + i) * 2 + 0];
    keys[i] = ((m >> 8) & 1) ? (__float_as_uint(n0) | 0x80000000u) : 0u;
  }
  __syncthreads();
  select_topk(keys, ind, K_POS, &cnt, scan);

  // negatives, noise channel 1
  for (int i = tid; i < NTOT; i += SEL_THREADS) {
    int m = match[(size_t)b * NTOT + i];
    float n1 = noise[((size_t)b * NTOT + i) * 2 + 1];
    keys[i] = ((m >> 9) & 1) ? (__float_as_uint(n1) | 0x80000000u) : 0u;
  }
  __syncthreads();
  select_topk(keys, ind, K_NEG, &cnt, scan);

  // stable top-512 of the 0/1 indicator = ones ascending, then zeros ascending
  int lo_i = tid * CHUNK;
  int hi_i = lo_i + CHUNK; if (hi_i > NTOT) hi_i = NTOT;
  int local = 0;
  for (int i = lo_i; i < hi_i; ++i) local += ind[i] ? 1 : 0;
  scan[tid] = local;
  __syncthreads();
  for (int off = 1; off < SEL_THREADS; off <<= 1) {
    int v = (tid >= off) ? scan[tid - off] : 0;
    __syncthreads();
    scan[tid] += v;
    __syncthreads();
  }
  const int total = scan[SEL_THREADS - 1];       // S = #selected (<= 512 always)
  int ones_before = scan[tid] - local;
  for (int i = lo_i; i < hi_i; ++i) {
    if (ind[i]) {
      if (ones_before < NUM_SAMPLED)
        sampled_idx[b * NUM_SAMPLED + ones_before] = i;
      ++ones_before;
    } else {
      int slot = total + (i - ones_before);      // zerosBefore(i) = i - onesBefore(i)
      if (slot < NUM_SAMPLED)
        sampled_idx[b * NUM_SAMPLED + slot] = i;
    }
  }
  if (tid == 0) num_ones[b] = total;
}

// ---------------- kernel 3: gather + delta encoding ----------------
#define O_ROIS 0
#define O_ENC  (BATCH * NUM_SAMPLED * 4)
#define O_BW   (2 * BATCH * NUM_SAMPLED * 4)
#define O_CLS  (O_BW + BATCH * NUM_SAMPLED)
#define O_CW   (O_CLS + BATCH * NUM_SAMPLED)

__global__ __launch_bounds__(256) void gather_kernel(
    const float* __restrict__ rois, const float* __restrict__ gt_boxes,
    const float* __restrict__ gt_classes, const int* __restrict__ match,
    const int* __restrict__ sampled_idx, const int* __restrict__ num_ones,
    float* __restrict__ out) {
  const int b = blockIdx.y;
  __shared__ float sgt[NUM_GT * 4];
  __shared__ float scls[NUM_GT];
#if USE_TDM
  if (threadIdx.x < 32) {
    tdm_load_dwords_to_lds((void*)sgt, (const void*)(gt_boxes + (size_t)b * NUM_GT * 4),
                           NUM_GT * 4);
    tdm_load_dwords_to_lds((void*)scls, (const void*)(gt_classes + (size_t)b * NUM_GT),
                           NUM_GT);
  }
  wait_tensorcnt0();
#else
  for (int t = threadIdx.x; t < NUM_GT * 4; t += blockDim.x)
    sgt[t] = gt_boxes[(size_t)b * NUM_GT * 4 + t];
  for (int t = threadIdx.x; t < NUM_GT; t += blockDim.x)
    scls[t] = gt_classes[(size_t)b * NUM_GT + t];
#endif
  __syncthreads();

  const int j = blockIdx.x * 256 + threadIdx.x;
  if (j >= NUM_SAMPLED) return;
  const int base = b * NUM_SAMPLED + j;
  const int idx = sampled_idx[base];
  const int m = match[(size_t)b * NTOT + idx];
  const int col = m & 0xFF;
  const bool pos = ((m >> 8) & 1) != 0;

  float a0, a1, a2, a3;
  if (idx < NUM_ROIS) {
    const float4* r4 = reinterpret_cast<const float4*>(rois);
    float4 v = r4[(size_t)b * NUM_ROIS + idx];
    a0 = v.x; a1 = v.y; a2 = v.z; a3 = v.w;
  } else {
    int g = idx - NUM_ROIS;
    a0 = sgt[g * 4 + 0]; a1 = sgt[g * 4 + 1]; a2 = sgt[g * 4 + 2]; a3 = sgt[g * 4 + 3];
  }
  out[O_ROIS + base * 4 + 0] = a0;
  out[O_ROIS + base * 4 + 1] = a1;
  out[O_ROIS + base * 4 + 2] = a2;
  out[O_ROIS + base * 4 + 3] = a3;

  float e0 = 0.f, e1 = 0.f, e2 = 0.f, e3 = 0.f;
  if (pos) {  // background (= !positive) -> zeros
    float b0 = sgt[col * 4 + 0], b1 = sgt[col * 4 + 1];
    float b2 = sgt[col * 4 + 2], b3 = sgt[col * 4 + 3];
    float ah = a2 - a0, aw = a3 - a1;
    float acy = a0 + 0.5f * ah, acx = a1 + 0.5f * aw;
    float bh = b2 - b0, bw = b3 - b1;
    float bcy = b0 + 0.5f * bh, bcx = b1 + 0.5f * bw;
    e0 = ((bcy - acy) / ah) / 0.1f;   // keep reference op order: delta then /VARIANCE
    e1 = ((bcx - acx) / aw) / 0.1f;
    e2 = logf(bh / ah) / 0.2f;
    e3 = logf(bw / aw) / 0.2f;
  }
  out[O_ENC + base * 4 + 0] = e0;
  out[O_ENC + base * 4 + 1] = e1;
  out[O_ENC + base * 4 + 2] = e2;
  out[O_ENC + base * 4 + 3] = e3;

  out[O_BW + base]  = pos ? 1.0f : 0.0f;            // sampled_box_weights
  out[O_CLS + base] = pos ? scls[col] : 0.0f;       // sampled_gt_classes (bg -> 0)
  out[O_CW + base]  = (j < num_ones[b]) ? 1.0f : 0.0f;  // sampled_class_weights
}

// ---------------- launch wrapper ----------------
extern "C" void kernel_launch(void* const* d_in, const int* in_sizes, int n_in,
                              void* d_out, int out_size, void* d_ws, size_t ws_size,
                              hipStream_t stream) {
  (void)in_sizes; (void)n_in; (void)out_size; (void)ws_size;
  const float* rois       = (const float*)d_in[0];  // (B, 8192, 4)
  const float* gt_boxes   = (const float*)d_in[1];  // (B, 128, 4)
  const float* gt_classes = (const float*)d_in[2];  // (B, 128, 1)
  const float* noise      = (const float*)d_in[3];  // (B, 8320, 2)
  float* out = (float*)d_out;

  int* match   = (int*)d_ws;                                  // B*NTOT ints
  int* sampled = match + (size_t)BATCH * NTOT;                // B*512 ints
  int* ones    = sampled + (size_t)BATCH * NUM_SAMPLED;       // B ints

  dim3 g1((NTOT + 255) / 256, BATCH);
  match_kernel<<<g1, 256, 0, stream>>>(rois, gt_boxes, match);

  sample_kernel<<<BATCH, SEL_THREADS, 0, stream>>>(noise, match, sampled, ones);

  dim3 g3((NUM_SAMPLED + 255) / 256, BATCH);
  gather_kernel<<<g3, 256, 0, stream>>>(rois, gt_boxes, gt_classes, match,
                                        sampled, ones, out);
}